// DepthWiseConv_36945308680631
// MI455X (gfx1250) — compile-verified
//
#include <hip/hip_runtime.h>
#include <stdint.h>

// Problem constants (reference: N=32, C=192, H=W=112, 3x3 depthwise, pad 1)
#define N_B        32
#define C_CH       192
#define HW_DIM     112
#define PLANE      (HW_DIM * HW_DIM)        // 12544 px per (n,c) plane
#define NPLANES    (N_B * C_CH)             // 6144
#define LDS_W      120                      // row stride (floats): 480B, 16B aligned
#define LDS_H      114                      // 112 data rows + 2 halo rows
#define COL0       4                        // data starts at col 4 (16B-aligned)
#define VEC_PER_ROW   28                    // 112 / 4
#define VEC_PER_PLANE (HW_DIM * VEC_PER_ROW) // 3136 float4 groups per plane
#define NTHREADS   256
// 256 = ROW_STEP*28 + C4_STEP  ->  per-iteration increment without div/mod
#define ROW_STEP   9
#define C4_STEP    4
#define BN_EPS     1e-5f

// ---- CDNA5 async global->LDS copy (per-lane, tracked by ASYNCcnt) ----------
__device__ __forceinline__ uint32_t lds_byte_addr(const void* p) {
  // addrspace(3) -> flat keeps the LDS byte offset in the low 32 bits
  return (uint32_t)(uintptr_t)p;
}

__device__ __forceinline__ void async_load_b128(uint32_t lds_addr, uint32_t goff,
                                                const float* sbase) {
  // GVS mode: mem = SGPR64 base + per-lane VGPR offset; dest = per-lane LDS addr
  asm volatile("global_load_async_to_lds_b128 %0, %1, %2"
               :: "v"(lds_addr), "v"(goff), "s"(sbase)
               : "memory");
}

__device__ __forceinline__ void wait_async0() {
  asm volatile("s_wait_asynccnt 0x0" ::: "memory");
}

// Stage one full (n,c) input plane into LDS with zeroed conv-padding halo.
__device__ __forceinline__ void load_plane_to_lds(float* tile, const float* gplane,
                                                  int t, int row0, int c40) {
  // Zero halo rows 0 and 113 (full width) and halo cols 3 and 116.
  for (int i = t; i < LDS_W; i += NTHREADS) {
    tile[i] = 0.0f;
    tile[113 * LDS_W + i] = 0.0f;
  }
  for (int r = 1 + t; r <= 112; r += NTHREADS) {
    tile[r * LDS_W + 3]   = 0.0f;
    tile[r * LDS_W + 116] = 0.0f;
  }
  const uint32_t base = lds_byte_addr(tile);
  int row = row0, c4 = c40;
  for (int v = t; v < VEC_PER_PLANE; v += NTHREADS) {
    const uint32_t laddr = base + (uint32_t)(((row + 1) * LDS_W + COL0 + 4 * c4) * 4);
    const uint32_t goff  = (uint32_t)((row * HW_DIM + 4 * c4) * 4);
    async_load_b128(laddr, goff, gplane);
    row += ROW_STEP; c4 += C4_STEP;
    if (c4 >= VEC_PER_ROW) { c4 -= VEC_PER_ROW; ++row; }
  }
  wait_async0();
  __syncthreads();
}

// 3x3 depthwise conv on a 1x4 output micro-tile read from the LDS plane.
__device__ __forceinline__ void conv4(const float* tile, int row, int c4,
                                      const float k[9], float acc[4]) {
  acc[0] = 0.0f; acc[1] = 0.0f; acc[2] = 0.0f; acc[3] = 0.0f;
#pragma unroll
  for (int rr = 0; rr < 3; ++rr) {
    const float* rp = &tile[(row + rr) * LDS_W + 3 + 4 * c4];
    const float x0 = rp[0], x1 = rp[1], x2 = rp[2];
    const float x3 = rp[3], x4 = rp[4], x5 = rp[5];
    const float k0 = k[rr * 3 + 0], k1 = k[rr * 3 + 1], k2 = k[rr * 3 + 2];
    acc[0] = fmaf(x2, k2, fmaf(x1, k1, fmaf(x0, k0, acc[0])));
    acc[1] = fmaf(x3, k2, fmaf(x2, k1, fmaf(x1, k0, acc[1])));
    acc[2] = fmaf(x4, k2, fmaf(x3, k1, fmaf(x2, k0, acc[2])));
    acc[3] = fmaf(x5, k2, fmaf(x4, k1, fmaf(x3, k0, acc[3])));
  }
}

// Pass 1: conv on the fly, per-plane partial sum / sumsq (fixed-order, deterministic)
__global__ __launch_bounds__(NTHREADS)
void dwconv_stats_kernel(const float* __restrict__ in, const float* __restrict__ w,
                         const float* __restrict__ b, float* __restrict__ psum,
                         float* __restrict__ psumsq) {
  __shared__ __align__(16) float tile[LDS_H * LDS_W];
  __shared__ float red[2 * NTHREADS];

  const int plane = blockIdx.x;
  const int c = plane % C_CH;
  const int t = threadIdx.x;
  const int row0 = t / VEC_PER_ROW;       // single div/mod per thread
  const int c40  = t % VEC_PER_ROW;
  const float* gplane = in + (size_t)plane * PLANE;

  float k[9];
#pragma unroll
  for (int i = 0; i < 9; ++i) k[i] = w[c * 9 + i];
  const float bias = b[c];

  load_plane_to_lds(tile, gplane, t, row0, c40);

  float s1 = 0.0f, s2 = 0.0f;
  int row = row0, c4 = c40;
  for (int v = t; v < VEC_PER_PLANE; v += NTHREADS) {
    float acc[4];
    conv4(tile, row, c4, k, acc);
#pragma unroll
    for (int j = 0; j < 4; ++j) {
      const float y = acc[j] + bias;
      s1 += y;
      s2 = fmaf(y, y, s2);
    }
    row += ROW_STEP; c4 += C4_STEP;
    if (c4 >= VEC_PER_ROW) { c4 -= VEC_PER_ROW; ++row; }
  }

  red[t] = s1;
  red[NTHREADS + t] = s2;
  __syncthreads();
#pragma unroll
  for (int off = NTHREADS / 2; off > 0; off >>= 1) {
    if (t < off) {
      red[t] += red[t + off];
      red[NTHREADS + t] += red[NTHREADS + t + off];
    }
    __syncthreads();
  }
  if (t == 0) {
    psum[plane]   = red[0];
    psumsq[plane] = red[NTHREADS];
  }
}

// Pass 2 (tiny): fold 32 per-n partials per channel -> scale/shift
__global__ void bn_finalize_kernel(const float* __restrict__ psum,
                                   const float* __restrict__ psumsq,
                                   const float* __restrict__ gamma,
                                   const float* __restrict__ beta,
                                   float* __restrict__ scale,
                                   float* __restrict__ shift) {
  const int c = blockIdx.x * blockDim.x + threadIdx.x;
  if (c >= C_CH) return;
  float s1 = 0.0f, s2 = 0.0f;
  for (int n = 0; n < N_B; ++n) {  // fixed order -> deterministic
    s1 += psum[n * C_CH + c];
    s2 += psumsq[n * C_CH + c];
  }
  const float invNHW = 1.0f / (float)((size_t)N_B * PLANE);
  const float mean = s1 * invNHW;
  const float var  = fmaf(-mean, mean, s2 * invNHW);
  const float inv  = rsqrtf(var + BN_EPS);
  const float sc   = gamma[c] * inv;
  scale[c] = sc;
  shift[c] = fmaf(-mean, sc, beta[c]);
}

// Pass 3: recompute conv (cheap), normalize, ReLU6, 128-bit stores
__global__ __launch_bounds__(NTHREADS)
void dwconv_apply_kernel(const float* __restrict__ in, const float* __restrict__ w,
                         const float* __restrict__ b, const float* __restrict__ scale,
                         const float* __restrict__ shift, float* __restrict__ out) {
  __shared__ __align__(16) float tile[LDS_H * LDS_W];

  const int plane = blockIdx.x;
  const int c = plane % C_CH;
  const int t = threadIdx.x;
  const int row0 = t / VEC_PER_ROW;
  const int c40  = t % VEC_PER_ROW;
  const float* gplane = in + (size_t)plane * PLANE;
  float* oplane = out + (size_t)plane * PLANE;

  float k[9];
#pragma unroll
  for (int i = 0; i < 9; ++i) k[i] = w[c * 9 + i];
  const float bias = b[c];
  const float sc = scale[c];
  const float sh = shift[c];

  load_plane_to_lds(tile, gplane, t, row0, c40);

  int row = row0, c4 = c40;
  for (int v = t; v < VEC_PER_PLANE; v += NTHREADS) {
    float acc[4];
    conv4(tile, row, c4, k, acc);
    float4 o;
    float z;
    z = fmaf(sc, acc[0] + bias, sh); o.x = fminf(fmaxf(z, 0.0f), 6.0f);
    z = fmaf(sc, acc[1] + bias, sh); o.y = fminf(fmaxf(z, 0.0f), 6.0f);
    z = fmaf(sc, acc[2] + bias, sh); o.z = fminf(fmaxf(z, 0.0f), 6.0f);
    z = fmaf(sc, acc[3] + bias, sh); o.w = fminf(fmaxf(z, 0.0f), 6.0f);
    *(float4*)(oplane + row * HW_DIM + 4 * c4) = o;  // 16B-aligned
    row += ROW_STEP; c4 += C4_STEP;
    if (c4 >= VEC_PER_ROW) { c4 -= VEC_PER_ROW; ++row; }
  }
}

extern "C" void kernel_launch(void* const* d_in, const int* in_sizes, int n_in,
                              void* d_out, int out_size, void* d_ws, size_t ws_size,
                              hipStream_t stream) {
  const float* in    = (const float*)d_in[0];  // (32,192,112,112)
  const float* w     = (const float*)d_in[1];  // (192,1,3,3)
  const float* b     = (const float*)d_in[2];  // (192,)
  const float* gamma = (const float*)d_in[3];  // (192,)
  const float* beta  = (const float*)d_in[4];  // (192,)
  float* out = (float*)d_out;

  float* ws     = (float*)d_ws;
  float* psum   = ws;                  // 6144
  float* psumsq = ws + NPLANES;        // 6144
  float* scale  = ws + 2 * NPLANES;    // 192
  float* shift  = ws + 2 * NPLANES + C_CH;  // 192  (~50 KB total, all overwritten each call)

  dwconv_stats_kernel<<<dim3(NPLANES), dim3(NTHREADS), 0, stream>>>(in, w, b, psum, psumsq);
  bn_finalize_kernel<<<dim3(1), dim3(C_CH), 0, stream>>>(psum, psumsq, gamma, beta, scale, shift);
  dwconv_apply_kernel<<<dim3(NPLANES), dim3(NTHREADS), 0, stream>>>(in, w, b, scale, shift, out);
}